// NTMReadHead_28492813042083
// MI455X (gfx1250) — compile-verified
//
#include <hip/hip_runtime.h>
#include <math.h>

// ---------------------------------------------------------------------------
// NTM read head for MI455X (gfx1250, wave32).
//  B=512, C=512, N=2048, M=64.  All fp32.
//  Kernel 1: o = emb @ fc_w^T + fc_b   via V_WMMA_F32_16X16X4_F32
//  Kernel 2: per-batch fused head (cosine sim -> softmax -> interp -> shift
//            -> sharpen -> normalize -> read vector), bandwidth-bound.
// ---------------------------------------------------------------------------

typedef __attribute__((ext_vector_type(2))) float v2f;
typedef __attribute__((ext_vector_type(8))) float v8f;

#define NTM_B 512
#define NTM_C 512
#define NTM_N 2048
#define NTM_M 64
#define NTM_J 70          // M + 6 output columns of the FC layer
#define NTM_EPS 1e-16f

// ---------------------------------------------------------------------------
// Kernel 1: FC GEMM with fp32 WMMA.
// One wave per 16x16 tile of o.  grid = (B/16, ceil(J/16)) = (32, 5).
// A-fragment (16x4 f32): lane%16 = M row, K = (lane/16)*2 + v  (v in 0..1)
// B-fragment (4x16 f32): lane%16 = N col, K = (lane/16)*2 + v
// C/D (16x16 f32, 8 VGPRs): row = r + (lane/16)*8, col = lane%16
// Out-of-range j columns are handled branch-free: clamp the row index to a
// valid one and scale the fragment by a 0/1 mask, so the K-loop has no
// exec-mask manipulation around the WMMA.
// ---------------------------------------------------------------------------
__global__ __launch_bounds__(32)
void ntm_fc_wmma_kernel(const float* __restrict__ emb,
                        const float* __restrict__ fcw,   // [J, C] row-major
                        const float* __restrict__ fcb,   // [J]
                        float* __restrict__ o)           // [B, J]
{
    const int lane = threadIdx.x;           // 0..31, single wave
    const int row  = lane & 15;
    const int half = lane >> 4;             // 0 or 1

    const int btile = blockIdx.x;           // 0..31
    const int jtile = blockIdx.y;           // 0..4

    // A rows: embeddings[btile*16 + row][*]
    const float2* arow =
        (const float2*)(emb + (size_t)(btile * 16 + row) * NTM_C);

    // B columns: fc_w^T[*][jtile*16 + row]  ==  fc_w[jtile*16 + row][*]
    const int   jcol   = jtile * 16 + row;
    const int   jclamp = (jcol < NTM_J) ? jcol : (NTM_J - 1);
    const float jmask  = (jcol < NTM_J) ? 1.0f : 0.0f;
    const float2* brow = (const float2*)(fcw + (size_t)jclamp * NTM_C);

    v8f acc = {};
    #pragma unroll 8
    for (int k = 0; k < NTM_C; k += 4) {
        // contiguous float2 at K = k + half*2
        float2 av = arow[(k >> 1) + half];
        v2f a; a[0] = av.x; a[1] = av.y;

        float2 bv = brow[(k >> 1) + half];      // unconditional load
        v2f b; b[0] = bv.x * jmask; b[1] = bv.y * jmask;

        acc = __builtin_amdgcn_wmma_f32_16x16x4_f32(
            /*neg_a=*/false, a, /*neg_b=*/false, b,
            /*c_mod=*/(short)0, acc, /*reuse_a=*/false, /*reuse_b=*/false);
    }

    // Store with fused bias.
    const int col = lane & 15;
    const int jg  = jtile * 16 + col;
    if (jg < NTM_J) {
        const float bias = fcb[jg];
        const int rbase = (lane >> 4) * 8;   // 0 or 8
        #pragma unroll
        for (int r = 0; r < 8; ++r) {
            const int mrow = btile * 16 + rbase + r;
            o[(size_t)mrow * NTM_J + jg] = acc[r] + bias;
        }
    }
}

// ---------------------------------------------------------------------------
// Kernel 2: fused per-batch head.  One block (256 threads = 8 waves) per b.
// ---------------------------------------------------------------------------
__global__ __launch_bounds__(256)
void ntm_head_kernel(const float* __restrict__ o,       // [B, J]
                     const float* __restrict__ w_prev,  // [B, N]
                     const float* __restrict__ memory,  // [B, N, M]
                     float* __restrict__ r_out,         // [B, M]
                     float* __restrict__ w_out)         // [B, N]
{
    __shared__ float zsh[NTM_N];     // z = beta*sim, then exp, then w_pow, then w
    __shared__ float wgsh[NTM_N];    // gated weights w_g
    __shared__ float ksh[NTM_M];     // k + EPS
    __shared__ float scal[8];        // 0:knorm 1:beta 2:g 3..5:s 6:gamma
    __shared__ float red[256];
    __shared__ float rpart[256];     // 4 chunks x 64 m

    const int b   = blockIdx.x;
    const int tid = threadIdx.x;
    const float* orow = o + (size_t)b * NTM_J;

    // ---------------- Stage A: head parameters -----------------------------
    if (tid < NTM_M) ksh[tid] = orow[tid] + NTM_EPS;
    __syncthreads();
    if (tid == 0) {
        float sq = 0.0f;
        #pragma unroll
        for (int i = 0; i < NTM_M; ++i) sq += ksh[i] * ksh[i];
        scal[0] = sqrtf(sq);                                   // ||k_e||
        scal[1] = fmaxf(orow[NTM_M + 0], 0.0f);                // beta = relu
        scal[2] = 1.0f / (1.0f + __expf(-orow[NTM_M + 1]));    // g = sigmoid
        float a0 = orow[NTM_M + 2], a1 = orow[NTM_M + 3], a2 = orow[NTM_M + 4];
        float mx = fmaxf(a0, fmaxf(a1, a2));
        float e0 = __expf(a0 - mx), e1 = __expf(a1 - mx), e2 = __expf(a2 - mx);
        float inv = 1.0f / (e0 + e1 + e2);
        scal[3] = e0 * inv; scal[4] = e1 * inv; scal[5] = e2 * inv;
        float x = orow[NTM_M + 5];                             // softplus, stable
        scal[6] = 1.0f + fmaxf(x, 0.0f) + log1pf(__expf(-fabsf(x)));
    }
    __syncthreads();

    const float knorm = scal[0];
    const float beta  = scal[1];
    const float g     = scal[2];
    const float s0 = scal[3], s1 = scal[4], s2 = scal[5];
    const float gamma = scal[6];

    // ---------------- Stage B: cosine similarity (memory pass 1) -----------
    // 16 threads per row n; each loads a float4 (4 of the 64 m-values).
    {
        const int sub = tid & 15;        // which float4 of the row
        const int grp = tid >> 4;        // 16 rows processed per iteration
        const float4 kv = ((const float4*)ksh)[sub];
        for (int it = 0; it < NTM_N / 16; ++it) {
            const int n = it * 16 + grp;
            const float4 mv = ((const float4*)(memory +
                ((size_t)b * NTM_N + n) * NTM_M))[sub];
            const float mx0 = mv.x + NTM_EPS, mx1 = mv.y + NTM_EPS;
            const float mx2 = mv.z + NTM_EPS, mx3 = mv.w + NTM_EPS;
            float dp = mx0 * kv.x + mx1 * kv.y + mx2 * kv.z + mx3 * kv.w;
            float np = mx0 * mx0 + mx1 * mx1 + mx2 * mx2 + mx3 * mx3;
            // reduce over the 16-lane subgroup (masks < 16 stay in-half)
            #pragma unroll
            for (int off = 8; off >= 1; off >>= 1) {
                dp += __shfl_xor(dp, off, 32);
                np += __shfl_xor(np, off, 32);
            }
            if (sub == 0) {
                const float denom = fmaxf(sqrtf(np) * knorm, 1e-8f);
                zsh[n] = beta * (dp / denom);
            }
        }
    }
    __syncthreads();

    // ---------------- Stage C: softmax / gate / shift / sharpen ------------
    // max over N
    float lmax = -INFINITY;
    for (int n = tid; n < NTM_N; n += 256) lmax = fmaxf(lmax, zsh[n]);
    red[tid] = lmax; __syncthreads();
    for (int s = 128; s > 0; s >>= 1) {
        if (tid < s) red[tid] = fmaxf(red[tid], red[tid + s]);
        __syncthreads();
    }
    const float zmax = red[0];
    __syncthreads();

    // exp + sum
    float lsum = 0.0f;
    for (int n = tid; n < NTM_N; n += 256) {
        const float e = __expf(zsh[n] - zmax);
        zsh[n] = e;
        lsum += e;
    }
    red[tid] = lsum; __syncthreads();
    for (int s = 128; s > 0; s >>= 1) {
        if (tid < s) red[tid] += red[tid + s];
        __syncthreads();
    }
    const float zsum = red[0];
    __syncthreads();

    // w_c -> w_g (gate with w_prev)
    const float invsum = 1.0f / zsum;
    const float* wp = w_prev + (size_t)b * NTM_N;
    for (int n = tid; n < NTM_N; n += 256) {
        const float wc = zsh[n] * invsum;
        wgsh[n] = g * wc + (1.0f - g) * wp[n];
    }
    __syncthreads();

    // circular shift + sharpen (w_s ^ gamma)
    float lpow = 0.0f;
    for (int n = tid; n < NTM_N; n += 256) {
        const int nm = (n + NTM_N - 1) & (NTM_N - 1);
        const int np = (n + 1) & (NTM_N - 1);
        const float ws = s0 * wgsh[nm] + s1 * wgsh[n] + s2 * wgsh[np];
        const float wpw = __powf(ws, gamma);   // ws >= 0
        zsh[n] = wpw;
        lpow += wpw;
    }
    red[tid] = lpow; __syncthreads();
    for (int s = 128; s > 0; s >>= 1) {
        if (tid < s) red[tid] += red[tid + s];
        __syncthreads();
    }
    const float psum = red[0];
    __syncthreads();

    const float invp = 1.0f / (psum + NTM_EPS);
    float* wrow = w_out + (size_t)b * NTM_N;
    for (int n = tid; n < NTM_N; n += 256) {
        const float w = zsh[n] * invp;
        zsh[n] = w;
        wrow[n] = w;
    }
    __syncthreads();

    // ---------------- Stage D: r = w . memory (memory pass 2) --------------
    // thread -> (chunk of N, m); fully coalesced 256B rows.
    {
        const int m     = tid & 63;
        const int chunk = tid >> 6;              // 0..3, 512 n each
        const float* mbase = memory + (size_t)b * NTM_N * NTM_M + m;
        float acc = 0.0f;
        const int n0 = chunk * (NTM_N / 4), n1 = n0 + (NTM_N / 4);
        for (int n = n0; n < n1; ++n)
            acc += zsh[n] * mbase[(size_t)n * NTM_M];
        rpart[tid] = acc;
    }
    __syncthreads();
    if (tid < NTM_M) {
        r_out[(size_t)b * NTM_M + tid] =
            rpart[tid] + rpart[64 + tid] + rpart[128 + tid] + rpart[192 + tid];
    }
}

// ---------------------------------------------------------------------------
extern "C" void kernel_launch(void* const* d_in, const int* in_sizes, int n_in,
                              void* d_out, int out_size, void* d_ws, size_t ws_size,
                              hipStream_t stream) {
    const float* emb    = (const float*)d_in[0];   // [512, 512]
    const float* w_prev = (const float*)d_in[1];   // [512, 2048]
    const float* memory = (const float*)d_in[2];   // [512, 2048, 64]
    const float* fc_w   = (const float*)d_in[3];   // [70, 512]
    const float* fc_b   = (const float*)d_in[4];   // [70]

    float* o_ws  = (float*)d_ws;                   // [512, 70] scratch
    float* r_out = (float*)d_out;                  // [512, 64]
    float* w_out = (float*)d_out + NTM_B * NTM_M;  // [512, 2048]

    dim3 g1(NTM_B / 16, (NTM_J + 15) / 16);        // (32, 5)
    ntm_fc_wmma_kernel<<<g1, 32, 0, stream>>>(emb, fc_w, fc_b, o_ws);

    ntm_head_kernel<<<NTM_B, 256, 0, stream>>>(o_ws, w_prev, memory,
                                               r_out, w_out);
}